// GraphCON_GCN_53128745451777
// MI455X (gfx1250) — compile-verified
//
#include <hip/hip_runtime.h>

#define N_NODES 100000
#define N_EDGES 1600000
#define HDIM    128
#define NGRAPHS 512
#define NLAYERS 5
#define DT_C    1.0f
#define ALPHA_C 1.0f
#define GAMMA_C 1.0f

typedef __attribute__((ext_vector_type(2))) float v2f;
typedef __attribute__((ext_vector_type(8))) float v8f;

// ---------------- degree / normalization ----------------
__global__ void k_init_deg(float* __restrict__ deg) {
    int i = blockIdx.x * blockDim.x + threadIdx.x;
    if (i < N_NODES) deg[i] = 1.0f;               // self-loop
}

__global__ void k_deg_edges(const int* __restrict__ dst, float* __restrict__ deg) {
    int e = blockIdx.x * blockDim.x + threadIdx.x;
    if (e < N_EDGES) atomicAdd(&deg[dst[e]], 1.0f);
}

__global__ void k_dinv(float* __restrict__ deg) {
    int i = blockIdx.x * blockDim.x + threadIdx.x;
    if (i < N_NODES) deg[i] = rsqrtf(deg[i]);     // in-place: deg -> dinv
}

// ---------------- encoder GEMM: Y = x @ W_enc^T + b; Xs = Y ----------------
// One wave per 16x16 output tile. K-loop in steps of 4 with V_WMMA_F32_16X16X4_F32.
// A 16x4 f32 layout: lane L holds A[M=L%16][K = 2*(L/16)+{0,1}]  (contiguous float2)
// B 4x16  f32 layout: lane L holds B[K = 2*(L/16)+{0,1}][N=L%16] = W[N][K...] (contiguous float2)
// C/D: VGPR v, lanes0-15 -> M=v, lanes16-31 -> M=v+8; N = lane%16.
__global__ void k_gemm_enc(const float* __restrict__ X, const float* __restrict__ W,
                           const float* __restrict__ b, float* __restrict__ Y,
                           float* __restrict__ Xs) {
    const int tile = blockIdx.x * 8 + (threadIdx.x >> 5);   // 8 waves / block
    const int mt = tile >> 3;          // 6250 row tiles
    const int nt = tile & 7;           // 8 col tiles (H=128)
    const int lane = threadIdx.x & 31;
    const int l15  = lane & 15;
    const int kh   = lane >> 4;        // 0/1
    const float* xrow = X + (size_t)(mt * 16 + l15) * HDIM + 2 * kh;
    const float* wrow = W + (size_t)(nt * 16 + l15) * HDIM + 2 * kh;

    v8f acc = {};
    #pragma unroll
    for (int k0 = 0; k0 < HDIM; k0 += 4) {
        v2f a  = *(const v2f*)(xrow + k0);
        v2f bv = *(const v2f*)(wrow + k0);
        acc = __builtin_amdgcn_wmma_f32_16x16x4_f32(false, a, false, bv,
                                                    (short)0, acc, false, false);
    }
    const int c = nt * 16 + l15;
    const float bias = b[c];
    #pragma unroll
    for (int v = 0; v < 8; ++v) {
        int r = mt * 16 + v + 8 * kh;
        size_t off = (size_t)r * HDIM + c;
        float val = acc[v] + bias;
        Y[off]  = val;
        Xs[off] = val;
    }
}

// ---------------- fused dual GEMM per layer ----------------
// xw = Xs @ W_conv^T
// S  = Xs @ W_res^T + b_res - xw + dinv^2 * xw + b_conv   (scatter target pre-init)
__global__ void k_gemm_dual(const float* __restrict__ Xs, const float* __restrict__ Wc,
                            const float* __restrict__ Wr, const float* __restrict__ bc,
                            const float* __restrict__ br, const float* __restrict__ dinv,
                            float* __restrict__ xw, float* __restrict__ S) {
    const int tile = blockIdx.x * 8 + (threadIdx.x >> 5);
    const int mt = tile >> 3;
    const int nt = tile & 7;
    const int lane = threadIdx.x & 31;
    const int l15  = lane & 15;
    const int kh   = lane >> 4;
    const float* xrow  = Xs + (size_t)(mt * 16 + l15) * HDIM + 2 * kh;
    const float* wcrow = Wc + (size_t)(nt * 16 + l15) * HDIM + 2 * kh;
    const float* wrrow = Wr + (size_t)(nt * 16 + l15) * HDIM + 2 * kh;

    v8f accC = {};
    v8f accR = {};
    #pragma unroll
    for (int k0 = 0; k0 < HDIM; k0 += 4) {
        v2f a  = *(const v2f*)(xrow + k0);
        v2f bC = *(const v2f*)(wcrow + k0);
        v2f bR = *(const v2f*)(wrrow + k0);
        accC = __builtin_amdgcn_wmma_f32_16x16x4_f32(false, a, false, bC,
                                                     (short)0, accC, false, false);
        accR = __builtin_amdgcn_wmma_f32_16x16x4_f32(false, a, false, bR,
                                                     (short)0, accR, false, false);
    }
    const int c = nt * 16 + l15;
    const float bconv = bc[c];
    const float bres  = br[c];
    #pragma unroll
    for (int v = 0; v < 8; ++v) {
        int r = mt * 16 + v + 8 * kh;
        size_t off = (size_t)r * HDIM + c;
        float conv = accC[v];
        float di = dinv[r];
        xw[off] = conv;
        S[off]  = accR[v] + bres - conv + di * di * conv + bconv;
    }
}

// ---------------- edge scatter: S[dst] += dinv[src]*dinv[dst] * xw[src] ----------------
__global__ void k_scatter(const int* __restrict__ src, const int* __restrict__ dst,
                          const float* __restrict__ dinv, const float* __restrict__ xw,
                          float* __restrict__ S) {
    long long idx = (long long)blockIdx.x * blockDim.x + threadIdx.x;
    int e = (int)(idx >> 5);                 // 32 threads per edge, 4 floats each
    if (e >= N_EDGES) return;
    int c = ((int)idx & 31) * 4;
    int s = src[e];
    int d = dst[e];
    float coef = dinv[s] * dinv[d];
    float4 v = *(const float4*)(xw + (size_t)s * HDIM + c);
    float* p = S + (size_t)d * HDIM + c;
    atomicAdd(p + 0, coef * v.x);
    atomicAdd(p + 1, coef * v.y);
    atomicAdd(p + 2, coef * v.z);
    atomicAdd(p + 3, coef * v.w);
}

// ---------------- GraphCON update ----------------
__global__ void k_update(const float* __restrict__ S, float* __restrict__ Y,
                         float* __restrict__ Xs) {
    size_t i = (size_t)blockIdx.x * blockDim.x + threadIdx.x;
    if (i >= (size_t)N_NODES * HDIM) return;
    float relu = fmaxf(S[i], 0.0f);
    float y  = Y[i];
    float xs = Xs[i];
    float yn = y + DT_C * (relu - ALPHA_C * y - GAMMA_C * xs);
    Y[i]  = yn;
    Xs[i] = xs + DT_C * yn;
}

// ---------------- decoder + segment-sum pooling ----------------
__global__ void k_zero_out(float* __restrict__ out) {
    int i = threadIdx.x;
    if (i < NGRAPHS) out[i] = 0.0f;
}

__global__ void k_dec_pool(const float* __restrict__ Xs, const float* __restrict__ Wd,
                           const float* __restrict__ bd, const int* __restrict__ batch,
                           float* __restrict__ out) {
    int n = blockIdx.x * blockDim.x + threadIdx.x;
    if (n >= N_NODES) return;
    const float4* xr = (const float4*)(Xs + (size_t)n * HDIM);
    const float4* wr = (const float4*)Wd;
    float acc = 0.0f;
    #pragma unroll
    for (int j = 0; j < HDIM / 4; ++j) {
        float4 a = xr[j];
        float4 w = wr[j];
        acc += a.x * w.x + a.y * w.y + a.z * w.z + a.w * w.w;
    }
    atomicAdd(&out[batch[n]], acc + bd[0]);
}

extern "C" void kernel_launch(void* const* d_in, const int* in_sizes, int n_in,
                              void* d_out, int out_size, void* d_ws, size_t ws_size,
                              hipStream_t stream) {
    const float* x      = (const float*)d_in[0];
    const int*   edge   = (const int*)d_in[1];   // [2, E]
    const int*   batch  = (const int*)d_in[2];
    const float* W_enc  = (const float*)d_in[3];
    const float* b_enc  = (const float*)d_in[4];
    const float* W_conv = (const float*)d_in[5];
    const float* b_conv = (const float*)d_in[6];
    const float* W_res  = (const float*)d_in[7];
    const float* b_res  = (const float*)d_in[8];
    const float* W_dec  = (const float*)d_in[9];
    const float* b_dec  = (const float*)d_in[10];

    const int* src = edge;
    const int* dst = edge + N_EDGES;

    float* ws   = (float*)d_ws;
    float* dinv = ws;                                  // N floats
    float* Y    = ws + N_NODES;                        // N*H
    float* Xs   = Y  + (size_t)N_NODES * HDIM;         // N*H
    float* xw   = Xs + (size_t)N_NODES * HDIM;         // N*H
    float* S    = xw + (size_t)N_NODES * HDIM;         // N*H

    k_init_deg <<<(N_NODES + 255) / 256, 256, 0, stream>>>(dinv);
    k_deg_edges<<<(N_EDGES + 255) / 256, 256, 0, stream>>>(dst, dinv);
    k_dinv     <<<(N_NODES + 255) / 256, 256, 0, stream>>>(dinv);

    const int tiles  = (N_NODES / 16) * (HDIM / 16);   // 50000, divisible by 8
    const int gblocks = tiles / 8;                     // 8 waves per 256-thread block

    k_gemm_enc<<<gblocks, 256, 0, stream>>>(x, W_enc, b_enc, Y, Xs);

    const long long sthreads = (long long)N_EDGES * (HDIM / 4) * 1;  // E * 32
    const unsigned  sblocks  = (unsigned)((sthreads + 255) / 256);
    const size_t    uthreads = (size_t)N_NODES * HDIM;
    const unsigned  ublocks  = (unsigned)((uthreads + 255) / 256);

    for (int l = 0; l < NLAYERS; ++l) {
        k_gemm_dual<<<gblocks, 256, 0, stream>>>(Xs, W_conv, W_res, b_conv, b_res,
                                                 dinv, xw, S);
        k_scatter  <<<sblocks, 256, 0, stream>>>(src, dst, dinv, xw, S);
        k_update   <<<ublocks, 256, 0, stream>>>(S, Y, Xs);
    }

    k_zero_out<<<1, 512, 0, stream>>>((float*)d_out);
    k_dec_pool<<<(N_NODES + 255) / 256, 256, 0, stream>>>(Xs, W_dec, b_dec, batch,
                                                          (float*)d_out);
}